// PIGNODE_17763984736723
// MI455X (gfx1250) — compile-verified
//
#include <hip/hip_runtime.h>
#include <math.h>

#define HID   64
#define HEADS 4
#define CCH   64
#define OUTW  256   // HEADS*CCH

typedef __attribute__((ext_vector_type(2))) float v2f;
typedef __attribute__((ext_vector_type(8))) float v8f;

// ---------------------------------------------------------------------------
// float atomic max via int/uint atomics (init with -inf bits 0xFF800000)
// ---------------------------------------------------------------------------
__device__ __forceinline__ void atomic_max_f32(float* addr, float value) {
  if (value >= 0.0f)
    atomicMax((int*)addr, __float_as_int(value));
  else
    atomicMin((unsigned int*)addr, __float_as_uint(value));
}

// ---------------------------------------------------------------------------
// Async global -> LDS copy of 16 bytes per lane (GLOBAL_LOAD_ASYNC_TO_LDS_B128,
// tracked by ASYNCcnt). VDST vgpr = per-lane LDS byte offset, VADDR = 64-bit
// global address. The LDS byte offset is derived by ptrtoint of the actual
// element pointer (low 32 bits of a generic pointer to a __shared__ object are
// its LDS byte offset) — this captures the shared array's address in the asm,
// so the compiler cannot fold the subsequent LDS loads to undef.
// NOTE: no "m" operands — AMDGPU inline asm cannot select memory operands.
// ---------------------------------------------------------------------------
__device__ __forceinline__ void async_copy_b128(float* lds_elem,
                                                const void* gaddr) {
#if defined(__gfx1250__)
  unsigned lds_off = (unsigned)(unsigned long long)lds_elem;
  asm volatile("global_load_async_to_lds_b128 %0, %1, off"
               :
               : "v"(lds_off), "v"(gaddr)
               : "memory");
#endif
}

__device__ __forceinline__ void wait_asynccnt0() {
#if defined(__gfx1250__)
  asm volatile("s_wait_asynccnt 0x0" ::: "memory");
#endif
}

// ---------------------------------------------------------------------------
// Tiny precompute: p_src[k,h] = sum_c W[k, h*64+c] * a_src[h,c]   (64x4)
//                  p_dst analog, qe[d,h] = sum_c We[d,h*64+c]*a_edge[h,c] (3x4)
// One block of 256 threads.
// ---------------------------------------------------------------------------
__global__ void precompute_kernel(const float* __restrict__ W,
                                  const float* __restrict__ We,
                                  const float* __restrict__ a_src,
                                  const float* __restrict__ a_dst,
                                  const float* __restrict__ a_edge,
                                  float* __restrict__ p_src,
                                  float* __restrict__ p_dst,
                                  float* __restrict__ qe) {
  int t = threadIdx.x;           // 0..255
  int k = t >> 2;                // 0..63
  int h = t & 3;                 // 0..3
  float s1 = 0.f, s2 = 0.f;
  for (int c = 0; c < CCH; ++c) {
    float w = W[k * OUTW + h * CCH + c];
    s1 += w * a_src[h * CCH + c];
    s2 += w * a_dst[h * CCH + c];
  }
  p_src[t] = s1;
  p_dst[t] = s2;
  if (t < 3 * HEADS) {
    int d = t >> 2;              // 0..2
    float s = 0.f;
    for (int c = 0; c < CCH; ++c)
      s += We[d * OUTW + h * CCH + c] * a_edge[h * CCH + c];
    qe[t] = s;
  }
}

// ---------------------------------------------------------------------------
// x = Z @ W : [N,64] x [64,256] -> [N,256] via V_WMMA_F32_16X16X4_F32.
// All 8 waves of a block share one 16-row A tile (mt is block-uniform), so the
// contiguous 4KB tile of Z is staged into LDS once per block with
// GLOBAL_LOAD_ASYNC_TO_LDS_B128 (ASYNCcnt), then WMMA A fragments come from
// LDS (ds_load_b64) and only the per-wave B fragments hit global memory.
// A layout (16x4 f32): lanes0-15 hold M=lane, v0=K0,v1=K1 ; lanes16-31 v0=K2,v1=K3
// B layout (4x16 f32): lanes0-15 hold N=lane, v0=K0,v1=K1 ; lanes16-31 v0=K2,v1=K3
// C/D (16x16 f32): vr = row (r + 8*half), col = n0 + (lane&15)
// ---------------------------------------------------------------------------
__global__ void __launch_bounds__(256)
gemm_x_wmma(const float* __restrict__ Z, const float* __restrict__ W,
            float* __restrict__ X, int nRows) {
  __shared__ float atile[16 * HID];        // 4KB: rows m0..m0+15 of Z
  int t    = threadIdx.x;                  // 0..255
  int wid  = blockIdx.x * 8 + (t >> 5);    // global wave id
  int lane = t & 31;
  int mt = wid >> 4;                       // block-uniform (8 waves per block,
  int nt = wid & 15;                       //  window of 16 is 8-aligned)
  int m0 = mt * 16;
  if (m0 >= nRows) return;                 // block-uniform: safe w.r.t. barrier
  int n0 = nt * 16;
  int half = lane >> 4;
  int l    = lane & 15;

  // stage A tile: contiguous 4KB from Z (row stride == HID), 16B per thread
  async_copy_b128(&atile[t * 4], (const void*)(Z + (size_t)m0 * HID + t * 4));
  wait_asynccnt0();
  __syncthreads();

  v8f acc = {};
  const float* arow = &atile[l * HID];
#pragma unroll
  for (int k0 = 0; k0 < HID; k0 += 4) {
    int ka = k0 + 2 * half;
    v2f a, b;
    a.x = arow[ka];
    a.y = arow[ka + 1];
    b.x = W[(size_t)ka * OUTW + n0 + l];
    b.y = W[(size_t)(ka + 1) * OUTW + n0 + l];
    acc = __builtin_amdgcn_wmma_f32_16x16x4_f32(false, a, false, b,
                                                (short)0, acc, false, false);
  }
  float* xo = X + (size_t)(m0 + 8 * half) * OUTW + n0 + l;
#pragma unroll
  for (int r = 0; r < 8; ++r)
    xo[(size_t)r * OUTW] = acc[r];
}

// ---------------------------------------------------------------------------
// alpha_s[n,h] = sum_k Z[n,k]*p_src[k,h] ; alpha_d analog.  (thread per (n,h))
// ---------------------------------------------------------------------------
__global__ void node_alphas(const float* __restrict__ Z,
                            const float* __restrict__ p_src,
                            const float* __restrict__ p_dst,
                            float* __restrict__ as, float* __restrict__ ad,
                            int nRows) {
  int t = blockIdx.x * blockDim.x + threadIdx.x;
  if (t >= nRows * HEADS) return;
  int n = t >> 2, h = t & 3;
  float s1 = 0.f, s2 = 0.f;
  const float* zr = Z + (size_t)n * HID;
  for (int k = 0; k < HID; ++k) {
    float z = zr[k];
    s1 += z * p_src[k * HEADS + h];
    s2 += z * p_dst[k * HEADS + h];
  }
  as[t] = s1;
  ad[t] = s2;
}

// ---------------------------------------------------------------------------
// Init: segmax = -inf, denom = 0, accum = 0
// ---------------------------------------------------------------------------
__global__ void init_ws_kernel(float* __restrict__ segmax,
                               float* __restrict__ denom,
                               float* __restrict__ accum,
                               int n4, int n64) {
  int t = blockIdx.x * blockDim.x + threadIdx.x;
  if (t < n4) {
    ((unsigned int*)segmax)[t] = 0xFF800000u;   // -inf
    denom[t] = 0.f;
  }
  if (t < n64) accum[t] = 0.f;
}

// ---------------------------------------------------------------------------
// Pass 1: logit[e,h] = leaky_relu(as[src]+ad[dst]+ea·qe, 0.2); seg-max per dst
// ---------------------------------------------------------------------------
__global__ void edge_pass1(const int* __restrict__ ei,
                           const float* __restrict__ ea,
                           const float* __restrict__ as,
                           const float* __restrict__ ad,
                           const float* __restrict__ qe,
                           float* __restrict__ logit,
                           float* __restrict__ segmax, int E_) {
  int e = blockIdx.x * blockDim.x + threadIdx.x;
  if (e >= E_) return;
  int src = ei[e], dst = ei[E_ + e];
  float d0 = ea[e * 3 + 0], d1 = ea[e * 3 + 1], d2 = ea[e * 3 + 2];
#pragma unroll
  for (int h = 0; h < HEADS; ++h) {
    float v = as[src * HEADS + h] + ad[dst * HEADS + h]
            + d0 * qe[0 * HEADS + h] + d1 * qe[1 * HEADS + h]
            + d2 * qe[2 * HEADS + h];
    v = v > 0.f ? v : 0.2f * v;
    logit[e * HEADS + h] = v;
    atomic_max_f32(&segmax[dst * HEADS + h], v);
  }
}

// ---------------------------------------------------------------------------
// Pass 2: ex = exp(logit - segmax[dst]); denom[dst] += ex (ex overwrites logit)
// ---------------------------------------------------------------------------
__global__ void edge_pass2(const int* __restrict__ ei,
                           float* __restrict__ logit_ex,     // in: logit, out: ex
                           const float* __restrict__ segmax,
                           float* __restrict__ denom, int E_) {
  int t = blockIdx.x * blockDim.x + threadIdx.x;
  if (t >= E_ * HEADS) return;
  int e = t >> 2, h = t & 3;
  int dst = ei[E_ + e];
  float v = __expf(logit_ex[t] - segmax[dst * HEADS + h]);
  logit_ex[t] = v;
  atomicAdd(&denom[dst * HEADS + h], v);
}

// ---------------------------------------------------------------------------
// Pass 3: per edge, 64 lanes (one per channel c):
//   m[c] = 0.25 * sum_h (ex[e,h]/(denom[dst,h]+1e-16)) * x[src, h*64+c]
//   accum[dst, c] += m[c]      (head-mean folded into weight)
// ---------------------------------------------------------------------------
__global__ void edge_messages(const int* __restrict__ ei,
                              const float* __restrict__ X,
                              const float* __restrict__ exbuf,
                              const float* __restrict__ denom,
                              float* __restrict__ accum, int E_) {
  int t = blockIdx.x * blockDim.x + threadIdx.x;
  int e = t >> 6;
  int c = t & 63;
  if (e >= E_) return;
  int src = ei[e], dst = ei[E_ + e];
  float w0 = exbuf[e * HEADS + 0] / (denom[dst * HEADS + 0] + 1e-16f);
  float w1 = exbuf[e * HEADS + 1] / (denom[dst * HEADS + 1] + 1e-16f);
  float w2 = exbuf[e * HEADS + 2] / (denom[dst * HEADS + 2] + 1e-16f);
  float w3 = exbuf[e * HEADS + 3] / (denom[dst * HEADS + 3] + 1e-16f);
  const float* xr = X + (size_t)src * OUTW;
  float m = 0.25f * (w0 * xr[c] + w1 * xr[CCH + c]
                   + w2 * xr[2 * CCH + c] + w3 * xr[3 * CCH + c]);
  atomicAdd(&accum[(size_t)dst * CCH + c], m);
}

// ---------------------------------------------------------------------------
// Finalize: per node (one wave32, 2 channels/lane): +bias, LayerNorm, SiLU
// ---------------------------------------------------------------------------
__global__ void finalize_node(const float* __restrict__ accum,
                              const float* __restrict__ bias,
                              const float* __restrict__ ln_g,
                              const float* __restrict__ ln_b,
                              float* __restrict__ zout, int nRows) {
  int wid  = (blockIdx.x * blockDim.x + threadIdx.x) >> 5;
  int lane = threadIdx.x & 31;
  if (wid >= nRows) return;
  float v0 = accum[(size_t)wid * CCH + lane]       + bias[lane];
  float v1 = accum[(size_t)wid * CCH + 32 + lane]  + bias[32 + lane];
  float s = v0 + v1;
#pragma unroll
  for (int m = 16; m >= 1; m >>= 1) s += __shfl_xor(s, m, 32);
  float mu = s * (1.0f / 64.0f);
  float d0 = v0 - mu, d1 = v1 - mu;
  float q = d0 * d0 + d1 * d1;
#pragma unroll
  for (int m = 16; m >= 1; m >>= 1) q += __shfl_xor(q, m, 32);
  float rstd = rsqrtf(q * (1.0f / 64.0f) + 1e-5f);
  float y0 = d0 * rstd * ln_g[lane]      + ln_b[lane];
  float y1 = d1 * rstd * ln_g[32 + lane] + ln_b[32 + lane];
  y0 = y0 / (1.0f + __expf(-y0));   // SiLU
  y1 = y1 / (1.0f + __expf(-y1));
  zout[(size_t)wid * CCH + lane]      = y0;
  zout[(size_t)wid * CCH + 32 + lane] = y1;
}

// ---------------------------------------------------------------------------
// Host-side: one GAT layer
// ---------------------------------------------------------------------------
struct LayerWS {
  float *x, *as, *ad, *segmax, *denom, *logit_ex, *accum, *psrc, *pdst, *qe;
};

static void run_layer(hipStream_t stream, int N, int E,
                      const float* z_in, const int* ei, const float* ea,
                      const float* W, const float* We,
                      const float* a_src, const float* a_dst, const float* a_edge,
                      const float* bias, const float* ln_g, const float* ln_b,
                      float* z_out, const LayerWS& w) {
  precompute_kernel<<<1, 256, 0, stream>>>(W, We, a_src, a_dst, a_edge,
                                           w.psrc, w.pdst, w.qe);
  // one wave per 16x16 tile: (N/16)*16 waves = N waves = N*32 threads
  gemm_x_wmma<<<(N * 32) / 256, 256, 0, stream>>>(z_in, W, w.x, N);
  node_alphas<<<(N * HEADS + 255) / 256, 256, 0, stream>>>(z_in, w.psrc, w.pdst,
                                                           w.as, w.ad, N);
  init_ws_kernel<<<(N * CCH + 255) / 256, 256, 0, stream>>>(w.segmax, w.denom,
                                                            w.accum, N * HEADS,
                                                            N * CCH);
  edge_pass1<<<(E + 255) / 256, 256, 0, stream>>>(ei, ea, w.as, w.ad, w.qe,
                                                  w.logit_ex, w.segmax, E);
  edge_pass2<<<(E * HEADS + 255) / 256, 256, 0, stream>>>(ei, w.logit_ex,
                                                          w.segmax, w.denom, E);
  edge_messages<<<(E * CCH + 255) / 256, 256, 0, stream>>>(ei, w.x, w.logit_ex,
                                                           w.denom, w.accum, E);
  finalize_node<<<(N * 32 + 255) / 256, 256, 0, stream>>>(w.accum, bias, ln_g,
                                                          ln_b, z_out, N);
}

extern "C" void kernel_launch(void* const* d_in, const int* in_sizes, int n_in,
                              void* d_out, int out_size, void* d_ws, size_t ws_size,
                              hipStream_t stream) {
  // setup_inputs order:
  // 0:t 1:h 2:edge_index 3:edge_attr 4:W1 5:We1 6:att_src1 7:att_dst1
  // 8:att_edge1 9:b1 10:ln_g1 11:ln_b1 12:W2 13:We2 14:att_src2 15:att_dst2
  // 16:att_edge2 17:b2 18:ln_g2 19:ln_b2
  const float* h    = (const float*)d_in[1];
  const int*   ei   = (const int*)d_in[2];
  const float* ea   = (const float*)d_in[3];
  const float* W1   = (const float*)d_in[4];
  const float* We1  = (const float*)d_in[5];
  const float* as1  = (const float*)d_in[6];
  const float* ad1  = (const float*)d_in[7];
  const float* ae1  = (const float*)d_in[8];
  const float* b1   = (const float*)d_in[9];
  const float* g1   = (const float*)d_in[10];
  const float* bb1  = (const float*)d_in[11];
  const float* W2   = (const float*)d_in[12];
  const float* We2  = (const float*)d_in[13];
  const float* as2  = (const float*)d_in[14];
  const float* ad2  = (const float*)d_in[15];
  const float* ae2  = (const float*)d_in[16];
  const float* b2   = (const float*)d_in[17];
  const float* g2   = (const float*)d_in[18];
  const float* bb2  = (const float*)d_in[19];

  const int N = in_sizes[1] / HID;   // 32768
  const int E = in_sizes[2] / 2;     // 262144

  float* ws = (float*)d_ws;
  LayerWS w;
  w.x        = ws;                       ws += (size_t)N * OUTW;   // 8M floats
  w.as       = ws;                       ws += (size_t)N * HEADS;
  w.ad       = ws;                       ws += (size_t)N * HEADS;
  w.segmax   = ws;                       ws += (size_t)N * HEADS;
  w.denom    = ws;                       ws += (size_t)N * HEADS;
  w.logit_ex = ws;                       ws += (size_t)E * HEADS;  // logit then ex
  w.accum    = ws;                       ws += (size_t)N * CCH;
  float* z1  = ws;                       ws += (size_t)N * CCH;
  w.psrc     = ws;                       ws += 256;
  w.pdst     = ws;                       ws += 256;
  w.qe       = ws;                       ws += 16;

  run_layer(stream, N, E, h,  ei, ea, W1, We1, as1, ad1, ae1, b1, g1, bb1, z1, w);
  run_layer(stream, N, E, z1, ei, ea, W2, We2, as2, ad2, ae2, b2, g2, bb2,
            (float*)d_out, w);
}